// MultiHeadAttention_82781199663476
// MI455X (gfx1250) — compile-verified
//
#include <hip/hip_runtime.h>
#include <math.h>

// ---------------------------------------------------------------------------
// Problem constants (from the reference)
// ---------------------------------------------------------------------------
#define B_      32
#define H_      8
#define N_      257
#define D_      64
#define DIM_    512
#define TRI_    1536      // 3*DIM
#define PIX_    256
#define NT_     17        // ceil(257/16)
#define SCALE_  0.125f    // D^-0.5
#define ISD_    0.125f    // 1/sqrt(D) for softmax temperature
#define EPS_    1e-5f

typedef __attribute__((ext_vector_type(2))) float v2f;
typedef __attribute__((ext_vector_type(8))) float v8f;

// f32 WMMA: D(16x16) = A(16x4) * B(4x16) + C   -> v_wmma_f32_16x16x4_f32
static __device__ __forceinline__ v8f wmma4(v2f a, v2f b, v8f c) {
  return __builtin_amdgcn_wmma_f32_16x16x4_f32(
      /*neg_a=*/false, a, /*neg_b=*/false, b,
      /*c_mod=*/(short)0, c, /*reuse_a=*/false, /*reuse_b=*/false);
}

static __device__ __forceinline__ unsigned hash_u32(unsigned x) {
  x ^= x >> 16; x *= 0x7feb352du;
  x ^= x >> 15; x *= 0x846ca68bu;
  x ^= x >> 16; return x;
}

// ---------------------------------------------------------------------------
// Kernel 1: per-batch mask vector m[b,256] and 1/(max m)^2
// ---------------------------------------------------------------------------
__global__ void mask_precompute_kernel(const int* __restrict__ msk,
                                       float* __restrict__ mvals,
                                       float* __restrict__ invmaxsq) {
  __shared__ float smax[PIX_];
  const int b = blockIdx.x;
  const int p = threadIdx.x;               // 0..255
  int v = msk[b * PIX_ + p];
  if (v == 0) v = 1 + (int)(hash_u32((unsigned)(b * PIX_ + p) * 0x9E3779B9u) % 127u);
  const float m = (float)v;
  mvals[b * PIX_ + p] = m;
  smax[p] = m;
  __syncthreads();
  for (int s = PIX_ / 2; s > 0; s >>= 1) {
    if (p < s) smax[p] = fmaxf(smax[p], smax[p + s]);
    __syncthreads();
  }
  if (p == 0) {
    const float mx = smax[0];
    invmaxsq[b] = 1.0f / (mx * mx);
  }
}

// ---------------------------------------------------------------------------
// Kernel 2: qkv = x @ W_qkv^T.   x:[8224,512]  W:[1536,512]  -> qkv:[8224,1536]
// One wave per 16x16 tile. Mtiles=514, Ntiles=96.
// ---------------------------------------------------------------------------
__global__ void qkv_gemm_kernel(const float* __restrict__ x,
                                const float* __restrict__ W,
                                float* __restrict__ qkv) {
  const int wid  = blockIdx.x * (blockDim.x >> 5) + (threadIdx.x >> 5);
  const int lane = threadIdx.x & 31;
  const int total = 514 * 96;
  if (wid >= total) return;
  const int nt = wid % 96, mt = wid / 96;
  const int m0 = mt * 16, n0 = nt * 16;
  const int half = lane >> 4, l15 = lane & 15;

  const float* arow = x + (size_t)(m0 + l15) * DIM_ + 2 * half;  // A[M=l15, K base]
  const float* brow = W + (size_t)(n0 + l15) * DIM_ + 2 * half;  // B[N=l15, K base] (W row-major = B^T)

  v8f c = {};
#pragma unroll 4
  for (int k = 0; k < DIM_; k += 4) {
    v2f a, bfrag;
    a.x = arow[k];     a.y = arow[k + 1];
    bfrag.x = brow[k]; bfrag.y = brow[k + 1];
    c = wmma4(a, bfrag, c);
  }
  // C/D layout: VGPR r -> row r + 8*half, col = l15
  float* orow = qkv + (size_t)(m0 + 8 * half) * TRI_ + n0 + l15;
#pragma unroll
  for (int r = 0; r < 8; ++r) orow[(size_t)r * TRI_] = c[r];
}

// ---------------------------------------------------------------------------
// Kernel 3: retention[b,h,i,j] = scale * sum_d q[b,i,h,d]*k[b,j,h,d]
// q/k live inside qkv buffer (offsets 0 / 512). One wave per 16x16 tile.
// ---------------------------------------------------------------------------
__global__ void retention_gemm_kernel(const float* __restrict__ qkv,
                                      float* __restrict__ ret) {
  const int wid  = blockIdx.x * (blockDim.x >> 5) + (threadIdx.x >> 5);
  const int lane = threadIdx.x & 31;
  const int total = B_ * H_ * NT_ * NT_;
  if (wid >= total) return;
  const int nt = wid % NT_;
  const int mt = (wid / NT_) % NT_;
  const int h  = (wid / (NT_ * NT_)) % H_;
  const int b  = wid / (NT_ * NT_ * H_);
  const int i0 = mt * 16, j0 = nt * 16;
  const int half = lane >> 4, l15 = lane & 15;

  const int ia = i0 + l15;                  // A row (query index)
  const int jb = j0 + l15;                  // B col (key index)
  const bool av = (ia < N_), bv = (jb < N_);
  const float* qrow = qkv + (size_t)(b * N_ + (av ? ia : 0)) * TRI_ + h * D_ + 2 * half;
  const float* krow = qkv + (size_t)(b * N_ + (bv ? jb : 0)) * TRI_ + DIM_ + h * D_ + 2 * half;

  v8f c = {};
#pragma unroll
  for (int k = 0; k < D_; k += 4) {
    v2f a, bf;
    a.x  = av ? qrow[k]     : 0.0f;
    a.y  = av ? qrow[k + 1] : 0.0f;
    bf.x = bv ? krow[k]     : 0.0f;
    bf.y = bv ? krow[k + 1] : 0.0f;
    c = wmma4(a, bf, c);
  }
  float* obase = ret + (size_t)((b * H_ + h) * N_) * N_;
  const int js = j0 + l15;
#pragma unroll
  for (int r = 0; r < 8; ++r) {
    const int is = i0 + r + 8 * half;
    if (is < N_ && js < N_) obase[(size_t)is * N_ + js] = SCALE_ * c[r];
  }
}

// ---------------------------------------------------------------------------
// Kernel 4: masked[b,h,i,j] = ret * mask(b,i,j)   (written into weights_after
// region pre-softmax; consumed by the inner GEMM, then softmaxed in place)
// ---------------------------------------------------------------------------
__global__ void mask_mul_kernel(const float* __restrict__ ret,
                                const float* __restrict__ mvals,
                                const float* __restrict__ invmaxsq,
                                float* __restrict__ dst) {
  const size_t idx = (size_t)blockIdx.x * blockDim.x + threadIdx.x;
  const size_t tot = (size_t)B_ * H_ * N_ * N_;
  if (idx >= tot) return;
  const int j = (int)(idx % N_);
  const int i = (int)((idx / N_) % N_);
  const int b = (int)(idx / ((size_t)N_ * N_ * H_));
  float mk;
  if (j > i)       mk = 0.0f;
  else if (i == j) mk = 1.0f;
  else if (j == 0) mk = __expf(-5.0f * (float)i / (float)(N_ - 1));
  else             mk = mvals[b * PIX_ + (i - 1)] * mvals[b * PIX_ + (j - 1)] * invmaxsq[b];
  dst[idx] = ret[idx] * mk;
}

// ---------------------------------------------------------------------------
// Kernel 5: row softmax of (src * 1/sqrt(D)) over last dim (len 257).
// One wave32 per row; 9 elements per lane.
// ---------------------------------------------------------------------------
__global__ void softmax_rows_kernel(const float* __restrict__ src,
                                    float* __restrict__ dst) {
  const int row  = blockIdx.x * (blockDim.x >> 5) + (threadIdx.x >> 5);
  const int lane = threadIdx.x & 31;
  const int nrows = B_ * H_ * N_;
  if (row >= nrows) return;
  const float* s = src + (size_t)row * N_;
  float* d       = dst + (size_t)row * N_;

  float v[9];
  float mx = -INFINITY;
#pragma unroll
  for (int t = 0; t < 9; ++t) {
    const int idx = lane + 32 * t;
    v[t] = (idx < N_) ? s[idx] * ISD_ : -INFINITY;
    mx = fmaxf(mx, v[t]);
  }
#pragma unroll
  for (int off = 16; off > 0; off >>= 1) mx = fmaxf(mx, __shfl_xor(mx, off, 32));

  float sum = 0.0f;
#pragma unroll
  for (int t = 0; t < 9; ++t) {
    v[t] = (lane + 32 * t < N_) ? __expf(v[t] - mx) : 0.0f;
    sum += v[t];
  }
#pragma unroll
  for (int off = 16; off > 0; off >>= 1) sum += __shfl_xor(sum, off, 32);
  const float inv = 1.0f / sum;
#pragma unroll
  for (int t = 0; t < 9; ++t) {
    const int idx = lane + 32 * t;
    if (idx < N_) d[idx] = v[t] * inv;
  }
}

// ---------------------------------------------------------------------------
// Kernel 6: r[b,h,i,d] = sum_j masked[b,h,i,j]*v[b,j,h,d]
//                      + scale * sum_e q[b,i,h,e]*past_kv[b,h,e,d]
// One wave per 16x16 tile. Mtiles=17 (i), Ntiles=4 (d).
// ---------------------------------------------------------------------------
__global__ void inner_cross_gemm_kernel(const float* __restrict__ masked,
                                        const float* __restrict__ qkv,
                                        const float* __restrict__ past_kv,
                                        float* __restrict__ rbuf) {
  const int wid  = blockIdx.x * (blockDim.x >> 5) + (threadIdx.x >> 5);
  const int lane = threadIdx.x & 31;
  const int total = B_ * H_ * NT_ * 4;
  if (wid >= total) return;
  const int dt = wid % 4;
  const int mt = (wid / 4) % NT_;
  const int h  = (wid / (4 * NT_)) % H_;
  const int b  = wid / (4 * NT_ * H_);
  const int i0 = mt * 16, d0 = dt * 16;
  const int half = lane >> 4, l15 = lane & 15;

  const int ia = i0 + l15;
  const bool av = (ia < N_);
  const float* mrow = masked + ((size_t)((b * H_ + h) * N_ + (av ? ia : 0))) * N_;
  const int dcol = d0 + l15;

  // ---- inner: K over j (0..256, padded to 260) ----
  v8f c1 = {};
  for (int j0 = 0; j0 < 260; j0 += 4) {
    const int ka = j0 + 2 * half;          // this lane's two K slots: ka, ka+1
    v2f a, bf;
    a.x  = (av && ka     < N_) ? mrow[ka]     : 0.0f;
    a.y  = (av && ka + 1 < N_) ? mrow[ka + 1] : 0.0f;
    bf.x = (ka     < N_) ? qkv[(size_t)(b * N_ + ka    ) * TRI_ + 2 * DIM_ + h * D_ + dcol] : 0.0f;
    bf.y = (ka + 1 < N_) ? qkv[(size_t)(b * N_ + ka + 1) * TRI_ + 2 * DIM_ + h * D_ + dcol] : 0.0f;
    c1 = wmma4(a, bf, c1);
  }

  // ---- cross: K over e (0..63) ----
  const float* qrow = qkv + (size_t)(b * N_ + (av ? ia : 0)) * TRI_ + h * D_ + 2 * half;
  const float* pk   = past_kv + (size_t)(b * H_ + h) * D_ * D_;
  v8f c2 = {};
#pragma unroll
  for (int e = 0; e < D_; e += 4) {
    const int ke = e + 2 * half;
    v2f a, bf;
    a.x  = av ? qrow[e]     : 0.0f;
    a.y  = av ? qrow[e + 1] : 0.0f;
    bf.x = pk[(size_t)ke * D_ + dcol];
    bf.y = pk[(size_t)(ke + 1) * D_ + dcol];
    c2 = wmma4(a, bf, c2);
  }

  float* obase = rbuf + ((size_t)((b * H_ + h) * N_)) * D_;
#pragma unroll
  for (int r = 0; r < 8; ++r) {
    const int is = i0 + r + 8 * half;
    if (is < N_) obase[(size_t)is * D_ + dcol] = c1[r] + SCALE_ * c2[r];
  }
}

// ---------------------------------------------------------------------------
// Kernel 7: per-(b,h) GroupNorm over (N,D), write transposed output [B,N,H*D]
// ---------------------------------------------------------------------------
__global__ void groupnorm_kernel(const float* __restrict__ rbuf,
                                 const float* __restrict__ gnw,
                                 const float* __restrict__ gnb,
                                 float* __restrict__ out) {
  __shared__ float ssum[256];
  __shared__ float ssq[256];
  const int bh = blockIdx.x;               // 0..255
  const int h = bh % H_, b = bh / H_;
  const int tid = threadIdx.x;
  const float* src = rbuf + (size_t)bh * N_ * D_;
  const int cnt = N_ * D_;                 // 16448

  float s = 0.0f, q = 0.0f;
  for (int i = tid; i < cnt; i += 256) {
    const float v = src[i];
    s += v; q += v * v;
  }
  ssum[tid] = s; ssq[tid] = q;
  __syncthreads();
  for (int st = 128; st > 0; st >>= 1) {
    if (tid < st) { ssum[tid] += ssum[tid + st]; ssq[tid] += ssq[tid + st]; }
    __syncthreads();
  }
  const float mean = ssum[0] / (float)cnt;
  const float var  = ssq[0] / (float)cnt - mean * mean;
  const float rstd = rsqrtf(var + EPS_);
  const float w = gnw[h], bi = gnb[h];

  for (int i = tid; i < cnt; i += 256) {
    const int n = i / D_, d = i % D_;
    out[(size_t)(b * N_ + n) * DIM_ + h * D_ + d] = (src[i] - mean) * rstd * w + bi;
  }
}

// ---------------------------------------------------------------------------
// Host launch
// ---------------------------------------------------------------------------
extern "C" void kernel_launch(void* const* d_in, const int* in_sizes, int n_in,
                              void* d_out, int out_size, void* d_ws, size_t ws_size,
                              hipStream_t stream) {
  const float* x       = (const float*)d_in[0];
  const int*   msk     = (const int*)d_in[1];
  const float* Wqkv    = (const float*)d_in[2];
  const float* past_kv = (const float*)d_in[3];
  const float* gnw     = (const float*)d_in[4];
  const float* gnb     = (const float*)d_in[5];

  float* out = (float*)d_out;                                   // [B,N,512]
  float* wb  = out + (size_t)B_ * N_ * DIM_;                    // weights_before
  float* wa  = wb + (size_t)B_ * H_ * N_ * N_;                  // weights_after

  float* ws   = (float*)d_ws;
  float* qkv  = ws;                                             // [B,N,1536]
  float* ret  = qkv + (size_t)B_ * N_ * TRI_;                   // [B,H,N,N]
  float* rbuf = ret + (size_t)B_ * H_ * N_ * N_;                // [B,H,N,D]
  float* mv   = rbuf + (size_t)B_ * H_ * N_ * D_;               // [B,256]
  float* imx  = mv + (size_t)B_ * PIX_;                         // [B]

  // 1) mask vectors
  mask_precompute_kernel<<<B_, 256, 0, stream>>>(msk, mv, imx);
  // 2) qkv projection (WMMA f32): 514*96 tiles, 8 waves/block
  qkv_gemm_kernel<<<(514 * 96) / 8, 256, 0, stream>>>(x, Wqkv, qkv);
  // 3) retention = q k^T * scale: 32*8*17*17 tiles
  retention_gemm_kernel<<<(B_ * H_ * NT_ * NT_) / 8, 256, 0, stream>>>(qkv, ret);
  // 4) masked retention -> weights_after region (pre-softmax)
  mask_mul_kernel<<<(B_ * H_ * N_ * N_) / 256, 256, 0, stream>>>(ret, mv, imx, wa);
  // 5) inner + cross retention GEMM (reads pre-softmax masked values)
  inner_cross_gemm_kernel<<<(B_ * H_ * NT_ * 4) / 8, 256, 0, stream>>>(wa, qkv, past_kv, rbuf);
  // 6) softmaxes: raw -> weights_before ; masked in-place -> weights_after
  softmax_rows_kernel<<<(B_ * H_ * N_) / 8, 256, 0, stream>>>(ret, wb);
  softmax_rows_kernel<<<(B_ * H_ * N_) / 8, 256, 0, stream>>>(wa, wa);
  // 7) GroupNorm + transpose to [B,N,H*D]
  groupnorm_kernel<<<B_ * H_, 256, 0, stream>>>(rbuf, gnw, gnb, out);
}